// BasicTT_39376260169960
// MI455X (gfx1250) — compile-verified
//
#include <hip/hip_runtime.h>
#include <math.h>

typedef float v2f __attribute__((ext_vector_type(2)));
typedef float v8f __attribute__((ext_vector_type(8)));
typedef int v4i __attribute__((vector_size(4 * sizeof(int))));
typedef __attribute__((address_space(1))) v4i* gv4i_ptr;
typedef __attribute__((address_space(3))) v4i* lv4i_ptr;

#define THREADS 256

// ---- CDNA5 async global->LDS copy (ASYNCcnt path), with sync fallback -----
#if defined(__has_builtin)
#if __has_builtin(__builtin_amdgcn_global_load_async_to_lds_b128)
#define USE_ASYNC_LDS 1
#endif
#endif
#ifndef USE_ASYNC_LDS
#define USE_ASYNC_LDS 0
#endif

__device__ __forceinline__ void async_wait_all() {
#if USE_ASYNC_LDS
#if defined(__has_builtin) && __has_builtin(__builtin_amdgcn_s_wait_asynccnt)
  __builtin_amdgcn_s_wait_asynccnt(0);
#else
  asm volatile("s_wait_asynccnt 0x0" ::: "memory");
#endif
#endif
}

// Stage the x-slice for one n4 into LDS: xs[k*64 + n3*8 + n5], 1536 floats,
// as 384 16-byte transfers (4 contiguous n5 values each).
__device__ __forceinline__ void stage_x(const float* __restrict__ xb, int n4,
                                        float* __restrict__ dst, int tid) {
  for (int q = tid; q < 384; q += THREADS) {
    int k = q >> 4;                 // 0..23  (n1,n2)
    int rem4 = (q & 15) << 2;       // 0..60  (n3*8 + n5base)
    int n3 = rem4 >> 3;
    int n5b = rem4 & 7;             // 0 or 4
    const float* g = xb + k * 512 + n3 * 64 + n4 * 8 + n5b;
#if USE_ASYNC_LDS
    __builtin_amdgcn_global_load_async_to_lds_b128(
        (gv4i_ptr)g, (lv4i_ptr)(dst + (q << 2)), 0, 0);
#else
    *(float4*)(dst + (q << 2)) = *(const float4*)g;
#endif
  }
}

// ---------------------------------------------------------------------------
// Setup kernel: precompute merged TT cores for layer 1 into d_ws.
//   GA  [64][24]  : (r2,m1,m2) x (n1,n2)   = sum_r1 l1c0*l1c1      @ ws[0]
//   A2  [16][16]  : (r3,m3)pad x (r2,n3)   = l1c2 (rows 8..15 = 0) @ ws[1536]
//   GCD [16][128] : (m4,m5)    x (r3,n4,n5)= sum_r4 l1c3*l1c4      @ ws[1792]
// ---------------------------------------------------------------------------
__global__ __launch_bounds__(THREADS) void tt_setup_kernel(
    const float* __restrict__ l1c0, const float* __restrict__ l1c1,
    const float* __restrict__ l1c2, const float* __restrict__ l1c3,
    const float* __restrict__ l1c4, float* __restrict__ ws) {
  const int tid = threadIdx.x;
  for (int idx = tid; idx < 1536; idx += THREADS) {        // GA
    int Mg = idx / 24, kk = idx % 24;
    int r2 = Mg >> 5, m1 = (Mg >> 2) & 7, m2 = Mg & 3;
    int n1 = kk / 8, n2 = kk & 7;
    float acc = 0.f;
    for (int r1 = 0; r1 < 3; ++r1)
      acc += l1c0[m1 * 9 + n1 * 3 + r1] *
             l1c1[((r1 * 4 + m2) * 8 + n2) * 2 + r2];
    ws[idx] = acc;
  }
  for (int idx = tid; idx < 256; idx += THREADS) {         // A2 (zero padded)
    int Mg = idx >> 4, kk = idx & 15;
    int r2 = kk >> 3, n3 = kk & 7;
    float v = 0.f;
    if (Mg < 8) {
      int r3 = Mg >> 2, m3 = Mg & 3;
      v = l1c2[((r2 * 4 + m3) * 8 + n3) * 2 + r3];
    }
    ws[1536 + idx] = v;
  }
  for (int idx = tid; idx < 2048; idx += THREADS) {        // GCD
    int Mg = idx >> 7, kk = idx & 127;
    int m4 = Mg >> 2, m5 = Mg & 3;
    int r3 = kk >> 6, n4 = (kk >> 3) & 7, n5 = kk & 7;
    float acc = 0.f;
    for (int r4 = 0; r4 < 2; ++r4)
      acc += l1c3[((r3 * 4 + m4) * 8 + n4) * 2 + r4] *
             l1c4[(r4 * 4 + m5) * 8 + n5];
    ws[1792 + idx] = acc;
  }
}

// Generic scalar TT step mirroring the reference einsum:
//   out[q][s][m] = sum_{r,n} z[r][n*s + s_i] * G[((r*m+mi)*n+ni)*r1 + q]
__device__ __forceinline__ void tt_step(const float* __restrict__ z,
                                        float* __restrict__ o,
                                        const float* __restrict__ G,
                                        int r0, int m, int n, int r1, int s,
                                        int tid) {
  const int sm = s * m, total = r1 * sm, ns = n * s;
  for (int idx = tid; idx < total; idx += THREADS) {
    int q = idx / sm, rem = idx - q * sm;
    int si = rem / m, mi = rem - si * m;
    float acc = 0.f;
    for (int r = 0; r < r0; ++r)
      for (int ni = 0; ni < n; ++ni)
        acc += z[r * ns + ni * s + si] * G[((r * m + mi) * n + ni) * r1 + q];
    o[idx] = acc;
  }
}

#define WMMA_F32(A, B, C) \
  __builtin_amdgcn_wmma_f32_16x16x4_f32(false, (A), false, (B), (short)0, (C), false, false)

// ---------------------------------------------------------------------------
// Fully fused forward: one workgroup (8 wave32) per sample.
// ---------------------------------------------------------------------------
__global__ __launch_bounds__(THREADS) void tt_forward_kernel(
    const float* __restrict__ x, const float* __restrict__ ws,
    const float* __restrict__ b1,
    const float* __restrict__ l2c0, const float* __restrict__ l2c1,
    const float* __restrict__ l2c2, const float* __restrict__ l2c3,
    const float* __restrict__ l2c4, const float* __restrict__ b2,
    const float* __restrict__ l3c0, const float* __restrict__ l3c1,
    const float* __restrict__ l3c2, const float* __restrict__ l3c3,
    const float* __restrict__ l3c4, const float* __restrict__ b3,
    const float* __restrict__ W, const float* __restrict__ bl,
    float* __restrict__ out) {
  __shared__ __align__(16) float GA_s[1536];   // [64][24]
  __shared__ __align__(16) float A2_s[256];    // [16][16]
  __shared__ __align__(16) float GCD_s[2048];  // [16][128]
  __shared__ __align__(16) float zA[4096];     // [(r2,n3)=16][(n5,m1,m2)=256]
  __shared__ __align__(16) float zB[2048];     // [(r3,n5)=16][(m1,m2,m3)=128]
  __shared__ __align__(16) float hbuf[2048];
  __shared__ __align__(16) float xs_t0[3072];  // x ping-pong in loop; t0 after

  float* const xs0 = xs_t0;
  float* const xs1 = xs_t0 + 1536;
  float* const t0 = xs_t0;  // reused by layers 2/3 after the n4 loop

  const int tid = threadIdx.x;
  const int wave = tid >> 5;
  const int lane = tid & 31;
  const int colL = lane & 15;   // N-column / A-row within tile
  const int khalf = lane >> 4;  // which K half this lane holds
  const int bi = blockIdx.x;
  const float* xb = x + (size_t)bi * 12288;

  for (int i = tid; i < 1536; i += THREADS) GA_s[i] = ws[i];
  for (int i = tid; i < 256; i += THREADS) A2_s[i] = ws[1536 + i];
  for (int i = tid; i < 2048; i += THREADS) GCD_s[i] = ws[1792 + i];
  stage_x(xb, 0, xs0, tid);  // kick off first x slice
  __syncthreads();

  // Hoisted loop-invariant A-operand fragments.
  const int mt = wave >> 1;  // step-A M tile owned by this wave
  v2f aA[6];
#pragma unroll
  for (int ksi = 0; ksi < 6; ++ksi) {
    int k0 = ksi * 4 + 2 * khalf;
    aA[ksi].x = GA_s[(mt * 16 + colL) * 24 + k0];
    aA[ksi].y = GA_s[(mt * 16 + colL) * 24 + k0 + 1];
  }
  v2f aB[4];
#pragma unroll
  for (int ksi = 0; ksi < 4; ++ksi) {
    int k0 = ksi * 4 + 2 * khalf;
    aB[ksi].x = A2_s[colL * 16 + k0];
    aB[ksi].y = A2_s[colL * 16 + k0 + 1];
  }

  v8f ccd = {0.f, 0.f, 0.f, 0.f, 0.f, 0.f, 0.f, 0.f};  // CD accumulator

  for (int n4 = 0; n4 < 8; ++n4) {
    const float* xs = (n4 & 1) ? xs1 : xs0;
    async_wait_all();  // this wave's pending async copies (current slice)
    __syncthreads();   // all waves' slice data + previous-iter buffer reads
    if (n4 < 7) stage_x(xb, n4 + 1, (n4 & 1) ? xs0 : xs1, tid);  // prefetch

    // ---- Step A: out[(r2,m1,m2)=64][(n3,n5)=64], K=(n1,n2)=24
    for (int j = 0; j < 2; ++j) {
      int nt = (wave * 2 + j) & 3;
      int cg = nt * 16 + colL;  // (n3,n5)
      v8f c = {0.f, 0.f, 0.f, 0.f, 0.f, 0.f, 0.f, 0.f};
#pragma unroll
      for (int ksi = 0; ksi < 6; ++ksi) {
        int k0 = ksi * 4 + 2 * khalf;
        v2f bb;
        bb.x = xs[k0 * 64 + cg];
        bb.y = xs[(k0 + 1) * 64 + cg];
        c = WMMA_F32(aA[ksi], bb, c);
      }
      int n3 = cg >> 3, n5 = cg & 7;
      int Mb = mt * 16 + 8 * khalf;  // first M row held by this lane half
      int base = ((Mb >> 5) * 8 + n3) * 256 + n5 * 32 + (Mb & 31);
      *(float4*)&zA[base] = make_float4(c[0], c[1], c[2], c[3]);
      *(float4*)&zA[base + 4] = make_float4(c[4], c[5], c[6], c[7]);
    }
    __syncthreads();

    // ---- Step B: out[(r3,m3)=8 pad16][(n5,m1,m2)=256], K=(r2,n3)=16
    for (int j = 0; j < 2; ++j) {
      int nn = (wave * 2 + j) * 16 + colL;  // (n5,m1,m2)
      v8f c = {0.f, 0.f, 0.f, 0.f, 0.f, 0.f, 0.f, 0.f};
#pragma unroll
      for (int ksi = 0; ksi < 4; ++ksi) {
        int k0 = ksi * 4 + 2 * khalf;
        v2f bb;
        bb.x = zA[k0 * 256 + nn];
        bb.y = zA[(k0 + 1) * 256 + nn];
        c = WMMA_F32(aB[ksi], bb, c);
      }
      if (khalf == 0) {  // rows 0..7 = valid (r3,m3); rows 8..15 are padding
        int n5 = nn >> 5, m1 = (nn >> 2) & 7, m2 = nn & 3;
        int cc = (m1 * 4 + m2) * 4;  // (m1,m2,m3=0)
        *(float4*)&zB[n5 * 128 + cc] = make_float4(c[0], c[1], c[2], c[3]);
        *(float4*)&zB[(8 + n5) * 128 + cc] = make_float4(c[4], c[5], c[6], c[7]);
      }
    }
    __syncthreads();

    // ---- Step CD partial: M=(m4,m5)=16, local K=(r3,n5)=16, N-tile = wave
#pragma unroll
    for (int ksi = 0; ksi < 4; ++ksi) {
      int k0 = ksi * 4 + 2 * khalf;
      int ga = colL * 128 + (k0 >> 3) * 64 + n4 * 8 + (k0 & 7);
      v2f a, bb;
      a.x = GCD_s[ga];
      a.y = GCD_s[ga + 1];
      bb.x = zB[k0 * 128 + wave * 16 + colL];
      bb.y = zB[(k0 + 1) * 128 + wave * 16 + colL];
      ccd = WMMA_F32(a, bb, ccd);
    }
    __syncthreads();
  }

  // h1 = relu(CD + b1); h flat index = (m1,m2,m3)*16 + (m4,m5)
#pragma unroll
  for (int i = 0; i < 8; ++i) {
    int Mg = i + 8 * khalf;     // (m4,m5)
    int Ng = wave * 16 + colL;  // (m1,m2,m3)
    int h = Ng * 16 + Mg;
    hbuf[h] = fmaxf(ccd[i] + b1[h], 0.f);
  }
  __syncthreads();

  // ---- Layer 2 (scalar TT steps, exact reference einsum order)
  tt_step(hbuf, t0, l2c0, 1, 4, 8, 2, 256, tid); __syncthreads();
  tt_step(t0, hbuf, l2c1, 2, 2, 4, 2, 256, tid); __syncthreads();
  tt_step(hbuf, t0, l2c2, 2, 2, 4, 2, 128, tid); __syncthreads();
  tt_step(t0, hbuf, l2c3, 2, 2, 4, 2, 64, tid);  __syncthreads();
  tt_step(hbuf, t0, l2c4, 2, 2, 4, 1, 32, tid);  __syncthreads();
  for (int i = tid; i < 64; i += THREADS) hbuf[i] = fmaxf(t0[i] + b2[i], 0.f);
  __syncthreads();

  // ---- Layer 3
  tt_step(hbuf, t0, l3c0, 1, 2, 4, 2, 16, tid); __syncthreads();
  tt_step(t0, hbuf, l3c1, 2, 2, 2, 2, 16, tid); __syncthreads();
  tt_step(hbuf, t0, l3c2, 2, 2, 2, 2, 16, tid); __syncthreads();
  tt_step(t0, hbuf, l3c3, 2, 2, 2, 2, 16, tid); __syncthreads();
  tt_step(hbuf, t0, l3c4, 2, 2, 2, 1, 16, tid); __syncthreads();
  for (int i = tid; i < 32; i += THREADS) hbuf[i] = fmaxf(t0[i] + b3[i], 0.f);
  __syncthreads();

  // ---- Head: logits + log_softmax
  if (tid == 0) {
    float l0 = bl[0], l1v = bl[1];
    for (int j = 0; j < 32; ++j) {
      l0 += W[j] * hbuf[j];
      l1v += W[32 + j] * hbuf[j];
    }
    float mx = fmaxf(l0, l1v);
    float lse = mx + logf(expf(l0 - mx) + expf(l1v - mx));
    out[bi * 2 + 0] = l0 - lse;
    out[bi * 2 + 1] = l1v - lse;
  }
}

extern "C" void kernel_launch(void* const* d_in, const int* in_sizes, int n_in,
                              void* d_out, int out_size, void* d_ws,
                              size_t ws_size, hipStream_t stream) {
  const float* x = (const float*)d_in[0];
  const float* l1c0 = (const float*)d_in[1];
  const float* l1c1 = (const float*)d_in[2];
  const float* l1c2 = (const float*)d_in[3];
  const float* l1c3 = (const float*)d_in[4];
  const float* l1c4 = (const float*)d_in[5];
  const float* b1 = (const float*)d_in[6];
  const float* l2c0 = (const float*)d_in[7];
  const float* l2c1 = (const float*)d_in[8];
  const float* l2c2 = (const float*)d_in[9];
  const float* l2c3 = (const float*)d_in[10];
  const float* l2c4 = (const float*)d_in[11];
  const float* b2 = (const float*)d_in[12];
  const float* l3c0 = (const float*)d_in[13];
  const float* l3c1 = (const float*)d_in[14];
  const float* l3c2 = (const float*)d_in[15];
  const float* l3c3 = (const float*)d_in[16];
  const float* l3c4 = (const float*)d_in[17];
  const float* b3 = (const float*)d_in[18];
  const float* W = (const float*)d_in[19];
  const float* bl = (const float*)d_in[20];
  float* ws = (float*)d_ws;
  float* out = (float*)d_out;

  const int nB = in_sizes[0] / 12288;  // B = 2048

  tt_setup_kernel<<<1, THREADS, 0, stream>>>(l1c0, l1c1, l1c2, l1c3, l1c4, ws);
  tt_forward_kernel<<<nB, THREADS, 0, stream>>>(
      x, ws, b1, l2c0, l2c1, l2c2, l2c3, l2c4, b2, l3c0, l3c1, l3c2, l3c3,
      l3c4, b3, W, bl, out);
}